// PlaceGNN_9010841387576
// MI455X (gfx1250) — compile-verified
//
#include <hip/hip_runtime.h>

// GCN 2-layer forward for MI455X (gfx1250, wave32).
// deg -> rsqrt; H1 = X@W1 (fp32 WMMA); edge scatter-add; epilogue(+self-loop,
// +bias, ReLU); H2 = A1@W2 (fp32 WMMA); edge scatter-add; epilogue.
// fp32 WMMA (V_WMMA_F32_16X16X4_F32): workload is scatter/L2-bound (~2.6 GB
// on-die traffic vs 8.2 GFLOP), so exact fp32 tensor ops are free.
// W is pre-packed into K-pair-interleaved layout so each lane's B fragment is
// a single aligned ds_load_b64 (no VGPR repacking), with pair-row stride
// padded to 32 floats mod 64 so a wave32 b64 LDS read covers all 64 banks.
// LDS staging uses GLOBAL_LOAD_ASYNC_TO_LDS_B128 (ASYNCcnt) when available.

typedef __attribute__((ext_vector_type(2))) float v2f;
typedef __attribute__((ext_vector_type(8))) float v8f;
typedef int v4i __attribute__((vector_size(16)));  // matches builtin param type

#define IN_DIM  256
#define HID_DIM 128
#define OUT_DIM 64

// pair-row stride (floats) for pair-packed W: N*2 columns + 32 pad floats
// => row-to-row LDS offset = 128 bytes (mod 256), conflict-free b64 reads.
#define SPK2_OF(N) ((N) * 2 + 32)

// ---------------- degree kernels ----------------
__global__ void k_deg_init(float* __restrict__ deg, int n) {
  int i = blockIdx.x * blockDim.x + threadIdx.x;
  if (i < n) deg[i] = 1.0f;  // self-loop contributes 1 to every target degree
}

__global__ void k_deg_accum(const int* __restrict__ col, float* __restrict__ deg, int E) {
  int e = blockIdx.x * blockDim.x + threadIdx.x;
  if (e < E) unsafeAtomicAdd(&deg[col[e]], 1.0f);  // global_atomic_add_f32
}

__global__ void k_deg_rsqrt(float* __restrict__ deg, int n) {
  int i = blockIdx.x * blockDim.x + threadIdx.x;
  if (i < n) deg[i] = rsqrtf(deg[i]);  // deg >= 1 always (self-loops)
}

__global__ void k_zero4(float4* __restrict__ p, size_t n4) {
  size_t i = (size_t)blockIdx.x * blockDim.x + threadIdx.x;
  if (i < n4) p[i] = make_float4(0.f, 0.f, 0.f, 0.f);
}

// ---------------- W pre-pack: Wpk[kp][n] = {W[2kp][n], W[2kp+1][n]} ----------
__global__ void k_pack_pairs(const float* __restrict__ W, float* __restrict__ Wpk,
                             int Khalf, int N, int spk2) {
  int idx = blockIdx.x * blockDim.x + threadIdx.x;
  if (idx >= Khalf * N) return;
  int kp = idx / N;
  int n  = idx % N;
  float2 v;
  v.x = W[(size_t)(2 * kp) * N + n];
  v.y = W[(size_t)(2 * kp + 1) * N + n];
  *(float2*)(Wpk + (size_t)kp * spk2 + 2 * n) = v;
}

// ---------------- fp32 WMMA GEMM: C[M,N] = A[M,K] @ B[K,N] ----------------
// 2 waves / block, one 16-row M-tile per wave, all N tiles in registers.
// Bpk = pair-packed W; staged to LDS per 64-row K-chunk as a raw copy
// (async-to-LDS when the builtin exists), consumed via ds_load_b64.
template <int N_DIM, int K_DIM>
__global__ __launch_bounds__(64) void k_gemm_wmma(const float* __restrict__ A,
                                                  const float* __restrict__ Bpk,
                                                  float* __restrict__ C, int M) {
  constexpr int KC   = 64;                  // K rows per LDS chunk
  constexpr int NT   = N_DIM / 16;          // 16-wide N tiles
  constexpr int SPK2 = SPK2_OF(N_DIM);      // floats per pair-row
  constexpr int CHUNK_F = (KC / 2) * SPK2;  // floats per LDS chunk
  static_assert(K_DIM % KC == 0, "K chunking");
  static_assert(CHUNK_F % (64 * 4) == 0, "fill divisibility");
  __shared__ float Bs[CHUNK_F];             // 36KB (N=128) / 20KB (N=64)

  const int lane  = threadIdx.x & 31;
  const int wave  = threadIdx.x >> 5;
  const int mtile = blockIdx.x * 2 + wave;
  const int m0    = mtile * 16;
  const bool active = (m0 < M);             // wave-uniform

  const int mrow  = lane & 15;              // A: M index / B,C: N index
  const int khalf = (lane >> 4) << 1;       // 0 (lanes 0-15) or 2 (lanes 16-31)

  v8f acc[NT] = {};

  for (int kc = 0; kc < K_DIM; kc += KC) {
    __syncthreads();  // previous chunk fully consumed
    const float* src = Bpk + (size_t)(kc >> 1) * SPK2;
#if defined(__has_builtin) && __has_builtin(__builtin_amdgcn_global_load_async_to_lds_b128)
    for (int i = threadIdx.x * 4; i < CHUNK_F; i += 64 * 4) {
      __builtin_amdgcn_global_load_async_to_lds_b128(
          (__attribute__((address_space(1))) v4i*)(src + i),
          (__attribute__((address_space(3))) v4i*)(Bs + i), 0, 0);
    }
    asm volatile("s_wait_asynccnt 0" ::: "memory");
#else
    for (int i = threadIdx.x * 4; i < CHUNK_F; i += 64 * 4) {
      *(float4*)(Bs + i) = *(const float4*)(src + i);
    }
#endif
    __syncthreads();

    if (active) {
      const float* arow = A + (size_t)(m0 + mrow) * K_DIM + kc;
#pragma unroll 4
      for (int k = 0; k < KC; k += 4) {
        // A fragment: 16x4 fp32 (ISA 7.12.2): lane pair K = k+khalf, k+khalf+1
        const float2 av = *(const float2*)(arow + k + khalf);
        v2f a;
        a[0] = av.x;
        a[1] = av.y;
        const int pr = (k + khalf) >> 1;    // pair-row inside chunk
#pragma unroll
        for (int nt = 0; nt < NT; ++nt) {
          // B fragment: one aligned b64 LDS read, no repacking
          const v2f b = *(const v2f*)(Bs + pr * SPK2 + 2 * (nt * 16 + mrow));
          acc[nt] = __builtin_amdgcn_wmma_f32_16x16x4_f32(
              false, a, false, b, (short)0, acc[nt], false, false);
        }
      }
    }
  }

  if (active) {
    const int moff = (lane >> 4) * 8;       // C/D: M = v + 8*(lane>=16)
    const int nn   = lane & 15;
#pragma unroll
    for (int nt = 0; nt < NT; ++nt) {
#pragma unroll
      for (int v = 0; v < 8; ++v) {
        C[(size_t)(m0 + moff + v) * N_DIM + nt * 16 + nn] = acc[nt][v];
      }
    }
  }
}

// ---------------- edge scatter: out[col] += h[row] * dis[row]*dis[col] -------
template <int DIM>
__global__ void k_scatter(const float* __restrict__ h, const int* __restrict__ row,
                          const int* __restrict__ col, const float* __restrict__ dis,
                          float* __restrict__ out, int E) {
  const int wid  = (int)(((size_t)blockIdx.x * blockDim.x + threadIdx.x) >> 5);
  const int lane = threadIdx.x & 31;
  if (wid >= E) return;
  const int r = row[wid];
  const int c = col[wid];
  const float norm = dis[r] * dis[c];
  constexpr int VEC = DIM / 32;  // 4 (hid) or 2 (out)
  const float* hp = h + (size_t)r * DIM + lane * VEC;
  float* op = out + (size_t)c * DIM + lane * VEC;
#pragma unroll
  for (int j = 0; j < VEC; ++j) unsafeAtomicAdd(op + j, hp[j] * norm);
}

// ---------------- epilogue: agg += h*dis^2 (self loop) + bias, optional ReLU -
template <int DIM, bool RELU>
__global__ void k_finish(float* __restrict__ agg, const float* __restrict__ h,
                         const float* __restrict__ dis, const float* __restrict__ bias,
                         int n) {
  const size_t idx = (size_t)blockIdx.x * blockDim.x + threadIdx.x;
  if (idx >= (size_t)n * DIM) return;
  const int i = (int)(idx / DIM);
  const int d = (int)(idx % DIM);
  const float s = dis[i];
  float v = agg[idx] + h[idx] * (s * s) + bias[d];
  agg[idx] = RELU ? fmaxf(v, 0.f) : v;
}

extern "C" void kernel_launch(void* const* d_in, const int* in_sizes, int n_in,
                              void* d_out, int out_size, void* d_ws, size_t ws_size,
                              hipStream_t stream) {
  const float* x  = (const float*)d_in[0];
  const int*   ei = (const int*)d_in[1];
  const float* W1 = (const float*)d_in[2];
  const float* b1 = (const float*)d_in[3];
  const float* W2 = (const float*)d_in[4];
  const float* b2 = (const float*)d_in[5];
  float* out = (float*)d_out;

  const int Nn = in_sizes[0] / IN_DIM;  // 100000
  const int E  = in_sizes[1] / 2;       // 1600000
  const int* row = ei;
  const int* col = ei + E;

  constexpr int SPK2_1 = SPK2_OF(HID_DIM);  // 288 floats / pair-row
  constexpr int SPK2_2 = SPK2_OF(OUT_DIM);  // 160 floats / pair-row
  const size_t w1p_f = (size_t)(IN_DIM / 2) * SPK2_1;   // 36864 floats
  const size_t w2p_f = (size_t)(HID_DIM / 2) * SPK2_2;  // 10240 floats

  // workspace layout (floats): deg/dis | h1 | a1 | h2 | W1 packed | W2 packed
  float* deg = (float*)d_ws;
  float* h1  = deg + Nn;
  float* a1  = h1 + (size_t)Nn * HID_DIM;
  float* h2  = a1 + (size_t)Nn * HID_DIM;
  float* w1p = h2 + (size_t)Nn * OUT_DIM;
  float* w2p = w1p + w1p_f;
  const size_t need =
      ((size_t)Nn * (1 + HID_DIM + HID_DIM + OUT_DIM) + w1p_f + w2p_f) * sizeof(float);
  if (ws_size < need) return;

  const int TB = 256;

  // --- symmetric normalization ---
  k_deg_init<<<(Nn + TB - 1) / TB, TB, 0, stream>>>(deg, Nn);
  k_deg_accum<<<(E + TB - 1) / TB, TB, 0, stream>>>(col, deg, E);
  k_deg_rsqrt<<<(Nn + TB - 1) / TB, TB, 0, stream>>>(deg, Nn);

  // --- pair-pack both weight matrices (tiny) ---
  {
    const int n1 = (IN_DIM / 2) * HID_DIM;
    const int n2 = (HID_DIM / 2) * OUT_DIM;
    k_pack_pairs<<<(n1 + TB - 1) / TB, TB, 0, stream>>>(W1, w1p, IN_DIM / 2, HID_DIM, SPK2_1);
    k_pack_pairs<<<(n2 + TB - 1) / TB, TB, 0, stream>>>(W2, w2p, HID_DIM / 2, OUT_DIM, SPK2_2);
  }

  const int mtiles = (Nn + 15) / 16;          // 6250
  const int gemm_blocks = (mtiles + 1) / 2;   // 3125
  const size_t edge_threads = (size_t)E * 32;

  // --- layer 1: 256 -> 128, ReLU ---
  {
    const size_t nz = (size_t)Nn * HID_DIM;
    k_zero4<<<(int)((nz / 4 + TB - 1) / TB), TB, 0, stream>>>((float4*)a1, nz / 4);
    k_gemm_wmma<HID_DIM, IN_DIM><<<gemm_blocks, 64, 0, stream>>>(x, w1p, h1, Nn);
    k_scatter<HID_DIM><<<(int)((edge_threads + TB - 1) / TB), TB, 0, stream>>>(
        h1, row, col, deg, a1, E);
    k_finish<HID_DIM, true><<<(int)((nz + TB - 1) / TB), TB, 0, stream>>>(
        a1, h1, deg, b1, Nn);
  }

  // --- layer 2: 128 -> 64 ---
  {
    const size_t nz = (size_t)Nn * OUT_DIM;
    k_zero4<<<(int)((nz / 4 + TB - 1) / TB), TB, 0, stream>>>((float4*)out, nz / 4);
    k_gemm_wmma<OUT_DIM, HID_DIM><<<gemm_blocks, 64, 0, stream>>>(a1, w2p, h2, Nn);
    k_scatter<OUT_DIM><<<(int)((edge_threads + TB - 1) / TB), TB, 0, stream>>>(
        h2, row, col, deg, out, E);
    k_finish<OUT_DIM, false><<<(int)((nz + TB - 1) / TB), TB, 0, stream>>>(
        out, h2, deg, b2, Nn);
  }
}